// BatchedMambaCore_15083925143606
// MI455X (gfx1250) — compile-verified
//
#include <hip/hip_runtime.h>
#include <hip/hip_bf16.h>

typedef __attribute__((ext_vector_type(2))) float    v2f;
typedef __attribute__((ext_vector_type(8))) float    v8f;
typedef __attribute__((ext_vector_type(4))) unsigned v4u;
typedef __attribute__((ext_vector_type(8))) int      v8i;
typedef __attribute__((ext_vector_type(4))) int      v4i;

#define D_MODEL 256
#define D_STATE 16
#define D_CONV  4
#define K_DIR   4
#define SEQ_L   1024
#define BATCH   8
#define D_INNER 512
#define DT_RANK 16
#define NCH     48            // DT_RANK + 2*D_STATE
#define LN_EPS  1e-5f
#define JCHUNK  32            // scan steps staged per LDS tile

__device__ __forceinline__ float silu_f(float v) {
  return v * __builtin_amdgcn_rcpf(1.0f + __expf(-v));
}
__device__ __forceinline__ float softplus_f(float v) {
  return (v > 20.0f) ? v : log1pf(__expf(v));
}

// direction permutation: scan index j -> sequence position p
__device__ __forceinline__ int perm_map(int k, int j) {
  switch (k) {
    case 0:  return j;
    case 1:  return SEQ_L - 1 - j;
    case 2:  return (j < SEQ_L / 2) ? (2 * j) : (2 * (j - SEQ_L / 2) + 1);   // evens-then-odds
    default: return (j < SEQ_L / 2) ? (2 * j + 1) : (2 * (j - SEQ_L / 2));   // odds-then-evens
  }
}

// ---------------------------------------------------------------------------
// TDM: 1-row tile of `nfloats` contiguous f32 -> contiguous LDS (D# per
// CDNA5 ISA).  Wave-uniform; call from one wave only.
// ---------------------------------------------------------------------------
#if __has_builtin(__builtin_amdgcn_tensor_load_to_lds)
#define HAVE_TDM 1
__device__ __forceinline__ void tdm_load_tile(const float* gsrc, float* lds_dst,
                                              int nfloats) {
  const unsigned long long ga = (unsigned long long)(size_t)gsrc;
  const unsigned lds_off = (unsigned)(size_t)lds_dst;   // low 32 bits = LDS byte addr

  v4u g0;
  g0[0] = 1u;                                                        // count=1, user D#
  g0[1] = (unsigned)__builtin_amdgcn_readfirstlane((int)lds_off);    // lds_addr
  g0[2] = (unsigned)__builtin_amdgcn_readfirstlane((int)(ga & 0xFFFFFFFFull));
  g0[3] = ((unsigned)__builtin_amdgcn_readfirstlane(
              (int)((ga >> 32) & 0x1FFFFFFull))) | (2u << 30);       // addr hi | type=2

  const unsigned nf = (unsigned)nfloats;
  v8i g1;
  g1[0] = 0x20000;                        // data_size=2 (4B); no mask/pad/iterate
  g1[1] = (int)((nf & 0xFFFFu) << 16);    // tensor_dim0[15:0]
  g1[2] = (int)((nf >> 16) | (1u << 16)); // tensor_dim0[31:16] | tensor_dim1=1
  g1[3] = (int)((nf & 0xFFFFu) << 16);    // tile_dim0
  g1[4] = 1;                              // tile_dim1=1, tile_dim2=0
  g1[5] = (int)nf;                        // tensor_dim0_stride[31:0]
  g1[6] = (int)((nf & 0xFFFFu) << 16);    // tensor_dim1_stride[15:0]
  g1[7] = 0;
  v4i g2 = {1, 0, 0, 0};                  // tensor_dim2=1 (unused: tile_dim2=0)
  v4i g3 = {0, 0, 0, 0};
#if defined(__clang_major__) && __clang_major__ >= 23
  v8i gx = {0, 0, 0, 0, 0, 0, 0, 0};
  __builtin_amdgcn_tensor_load_to_lds(g0, g1, g2, g3, gx, 0);
#else
  __builtin_amdgcn_tensor_load_to_lds(g0, g1, g2, g3, 0);
#endif
}
#else
#define HAVE_TDM 0
#endif

// ---------------------------------------------------------------------------
// Kernel 1: xz = x @ in_proj_w^T ; split -> x_ssm, silu(z)
// M = B*L = 8192, N = 2*D_INNER = 1024, K = 256.  One 16x16 C-tile per wave.
// ---------------------------------------------------------------------------
__global__ void __launch_bounds__(256) k_in_proj(const float* __restrict__ x,
                                                 const float* __restrict__ w,
                                                 float* __restrict__ x_ssm,
                                                 float* __restrict__ z) {
  const int lane = threadIdx.x & 31;
  const int wave = threadIdx.x >> 5;
  const int tile = blockIdx.x * 8 + wave;      // 0..32767
  const int tM = tile >> 6;                    // 0..511
  const int tN = tile & 63;                    // 0..63
  const int lm = lane & 15;
  const int kh = (lane >> 4) << 1;             // K half-select per ISA A/B layout

  const float* arow = x + (size_t)(tM * 16 + lm) * D_MODEL + kh;
  const float* brow = w + (size_t)(tN * 16 + lm) * D_MODEL + kh;

  v8f acc = {0.f, 0.f, 0.f, 0.f, 0.f, 0.f, 0.f, 0.f};
#pragma unroll 8
  for (int k0 = 0; k0 < D_MODEL; k0 += 4) {
    if ((k0 & 31) == 0 && k0 + 32 < D_MODEL) {
      __builtin_prefetch(arow + k0 + 32, 0, 3);
      __builtin_prefetch(brow + k0 + 32, 0, 3);
    }
    v2f a = *(const v2f*)(arow + k0);
    v2f b = *(const v2f*)(brow + k0);
    acc = __builtin_amdgcn_wmma_f32_16x16x4_f32(false, a, false, b, (short)0, acc, false, false);
  }

  const int n = tN * 16 + lm;
  const int mbase = tM * 16 + ((lane >> 4) << 3);
#pragma unroll
  for (int v = 0; v < 8; ++v) {
    const int m = mbase + v;
    if (n < D_INNER) x_ssm[(size_t)m * D_INNER + n] = acc[v];
    else             z[(size_t)m * D_INNER + (n - D_INNER)] = silu_f(acc[v]);
  }
}

// ---------------------------------------------------------------------------
// Kernel 2: depthwise conv (pad 1,2) + bias + silu ; layout (b,l,d)
// ---------------------------------------------------------------------------
__global__ void __launch_bounds__(256) k_conv(const float* __restrict__ x_ssm,
                                              const float* __restrict__ cw,
                                              const float* __restrict__ cb,
                                              float* __restrict__ xc) {
  const int idx = blockIdx.x * 256 + threadIdx.x;   // b*L*D total
  const int d = idx & (D_INNER - 1);
  const int l = (idx >> 9) & (SEQ_L - 1);
  const int b = idx >> 19;
  float acc = cb[d];
#pragma unroll
  for (int j = 0; j < D_CONV; ++j) {
    const int li = l - 1 + j;
    if (li >= 0 && li < SEQ_L)
      acc += cw[d * D_CONV + j] * x_ssm[((size_t)b * SEQ_L + li) * D_INNER + d];
  }
  xc[((size_t)b * SEQ_L + l) * D_INNER + d] = silu_f(acc);
}

// ---------------------------------------------------------------------------
// Kernel 3: x_dbl[b,k,j,c] = sum_d x_proj_w[k,c,d] * xc[b, perm_k(j), d]
// Per (b,k): M = L (j), N = 48 (c), K = 512 (d).  WMMA f32 16x16x4.
// ---------------------------------------------------------------------------
__global__ void __launch_bounds__(256) k_xdbl(const float* __restrict__ xc,
                                              const float* __restrict__ xpw,
                                              float* __restrict__ x_dbl) {
  const int lane = threadIdx.x & 31;
  const int wave = threadIdx.x >> 5;
  int tile = blockIdx.x * 8 + wave;            // 0..6143
  const int ct = tile % 3;  tile /= 3;         // 0..2  (c tile)
  const int jt = tile & 63; tile >>= 6;        // 0..63 (j tile)
  const int k  = tile & 3;
  const int b  = tile >> 2;

  const int lm = lane & 15;
  const int kh = (lane >> 4) << 1;
  const int j  = jt * 16 + lm;
  const int p  = perm_map(k, j);

  const float* arow = xc  + ((size_t)b * SEQ_L + p) * D_INNER + kh;
  const float* brow = xpw + (size_t)(k * NCH + ct * 16 + lm) * D_INNER + kh;

  v8f acc = {0.f, 0.f, 0.f, 0.f, 0.f, 0.f, 0.f, 0.f};
#pragma unroll 8
  for (int k0 = 0; k0 < D_INNER; k0 += 4) {
    v2f a  = *(const v2f*)(arow + k0);
    v2f bb = *(const v2f*)(brow + k0);
    acc = __builtin_amdgcn_wmma_f32_16x16x4_f32(false, a, false, bb, (short)0, acc, false, false);
  }

  const int c  = ct * 16 + lm;
  const int jb = jt * 16 + ((lane >> 4) << 3);
#pragma unroll
  for (int v = 0; v < 8; ++v)
    x_dbl[(((size_t)(b * K_DIR + k) * SEQ_L) + (jb + v)) * NCH + c] = acc[v];
}

// ---------------------------------------------------------------------------
// Kernel 4: selective scan.  One block per (b,k); thread = channel d.
// h[16], A-row, dtW-row register-resident; dts/B/C columns staged in LDS via
// TDM (double-buffered: tensor_load of chunk c+1 overlaps compute of chunk c).
// ---------------------------------------------------------------------------
__global__ void __launch_bounds__(512) k_scan(const float* __restrict__ xc,
                                              const float* __restrict__ x_dbl,
                                              const float* __restrict__ dtw,
                                              const float* __restrict__ dtb,
                                              const float* __restrict__ alogs,
                                              const float* __restrict__ dvec,
                                              float* __restrict__ ys) {
  __shared__ float sd[2][JCHUNK * NCH];
  const int d  = threadIdx.x;
  const int bk = blockIdx.x;
  const int b = bk >> 2, k = bk & 3;

  float aR[D_STATE], wR[DT_RANK], h[D_STATE];
#pragma unroll
  for (int n = 0; n < D_STATE; ++n) {
    aR[n] = -__expf(alogs[(size_t)(k * D_INNER + d) * D_STATE + n]);
    wR[n] = dtw[(size_t)(k * D_INNER + d) * DT_RANK + n];
    h[n] = 0.f;
  }
  const float bias = dtb[k * D_INNER + d];
  const float Dd   = dvec[k * D_INNER + d];
  const float* xdb = x_dbl + (size_t)bk * SEQ_L * NCH;

#if HAVE_TDM
  if (threadIdx.x < 32)                      // wave 0 issues the DMA
    tdm_load_tile(xdb, sd[0], JCHUNK * NCH);

  for (int jc = 0; jc < SEQ_L; jc += JCHUNK) {
    const int buf = (jc / JCHUNK) & 1;
    if (threadIdx.x < 32)
      __builtin_amdgcn_s_wait_tensorcnt(0);  // tile for this chunk landed in LDS
    __syncthreads();
    if (threadIdx.x < 32 && jc + JCHUNK < SEQ_L)
      tdm_load_tile(xdb + (size_t)(jc + JCHUNK) * NCH, sd[buf ^ 1], JCHUNK * NCH);
#else
  for (int jc = 0; jc < SEQ_L; jc += JCHUNK) {
    const int buf = 0;
    __syncthreads();
    for (int t = threadIdx.x; t < JCHUNK * NCH; t += 512)
      sd[0][t] = xdb[(size_t)jc * NCH + t];
    __syncthreads();
#endif
    for (int jj = 0; jj < JCHUNK; ++jj) {
      const float* col = sd[buf] + jj * NCH;
      float dv = bias;
#pragma unroll
      for (int r = 0; r < DT_RANK; ++r) dv += wR[r] * col[r];
      const float delta = softplus_f(dv);
      const int p = perm_map(k, jc + jj);
      const float u  = xc[((size_t)b * SEQ_L + p) * D_INNER + d];
      const float du = delta * u;
      float acc = 0.f;
#pragma unroll
      for (int n = 0; n < D_STATE; ++n) {
        const float dA = __expf(delta * aR[n]);
        h[n] = h[n] * dA + du * col[DT_RANK + n];
        acc += h[n] * col[DT_RANK + D_STATE + n];
      }
      ys[((size_t)bk * SEQ_L + p) * D_INNER + d] = acc + Dd * u;
    }
  }
}

// ---------------------------------------------------------------------------
// Kernel 5: combine 4 directions + LayerNorm(512) + * silu(z)
// ---------------------------------------------------------------------------
__global__ void __launch_bounds__(256) k_norm(const float* __restrict__ ys,
                                              const float* __restrict__ z,
                                              const float* __restrict__ g,
                                              const float* __restrict__ be,
                                              float* __restrict__ yf) {
  __shared__ float s1[256], s2[256];
  const int row = blockIdx.x;                 // b*L + l
  const int b = row >> 10, l = row & (SEQ_L - 1);
  const int t = threadIdx.x;
  float v0 = 0.f, v1 = 0.f;
#pragma unroll
  for (int k = 0; k < K_DIR; ++k) {
    const float* base = ys + (((size_t)(b * K_DIR + k) * SEQ_L) + l) * D_INNER;
    v0 += base[t];
    v1 += base[t + 256];
  }
  s1[t] = v0 + v1;
  s2[t] = v0 * v0 + v1 * v1;
  __syncthreads();
  for (int o = 128; o > 0; o >>= 1) {
    if (t < o) { s1[t] += s1[t + o]; s2[t] += s2[t + o]; }
    __syncthreads();
  }
  const float mu  = s1[0] * (1.0f / 512.0f);
  const float var = s2[0] * (1.0f / 512.0f) - mu * mu;
  const float rs  = rsqrtf(var + LN_EPS);
  const float* zr = z + (size_t)row * D_INNER;
  float* yr = yf + (size_t)row * D_INNER;
  yr[t]       = ((v0 - mu) * rs * g[t]       + be[t])       * zr[t];
  yr[t + 256] = ((v1 - mu) * rs * g[t + 256] + be[t + 256]) * zr[t + 256];
}

// ---------------------------------------------------------------------------
// Kernel 6: out = y_final (8192x512) @ out_proj_w^T (512x256)
// ---------------------------------------------------------------------------
__global__ void __launch_bounds__(256) k_out_proj(const float* __restrict__ yf,
                                                  const float* __restrict__ w,
                                                  float* __restrict__ out) {
  const int lane = threadIdx.x & 31;
  const int wave = threadIdx.x >> 5;
  const int tile = blockIdx.x * 8 + wave;     // 0..8191
  const int tM = tile >> 4;                   // 0..511
  const int tN = tile & 15;                   // 0..15
  const int lm = lane & 15;
  const int kh = (lane >> 4) << 1;

  const float* arow = yf + (size_t)(tM * 16 + lm) * D_INNER + kh;
  const float* brow = w  + (size_t)(tN * 16 + lm) * D_INNER + kh;

  v8f acc = {0.f, 0.f, 0.f, 0.f, 0.f, 0.f, 0.f, 0.f};
#pragma unroll 8
  for (int k0 = 0; k0 < D_INNER; k0 += 4) {
    if ((k0 & 31) == 0 && k0 + 32 < D_INNER) {
      __builtin_prefetch(arow + k0 + 32, 0, 3);
      __builtin_prefetch(brow + k0 + 32, 0, 3);
    }
    v2f a = *(const v2f*)(arow + k0);
    v2f b = *(const v2f*)(brow + k0);
    acc = __builtin_amdgcn_wmma_f32_16x16x4_f32(false, a, false, b, (short)0, acc, false, false);
  }

  const int n = tN * 16 + lm;
  const int mbase = tM * 16 + ((lane >> 4) << 3);
#pragma unroll
  for (int v = 0; v < 8; ++v)
    out[(size_t)(mbase + v) * D_MODEL + n] = acc[v];
}

// ---------------------------------------------------------------------------
extern "C" void kernel_launch(void* const* d_in, const int* in_sizes, int n_in,
                              void* d_out, int out_size, void* d_ws, size_t ws_size,
                              hipStream_t stream) {
  const float* x          = (const float*)d_in[0];
  const float* in_proj_w  = (const float*)d_in[1];
  const float* conv_w     = (const float*)d_in[2];
  const float* conv_b     = (const float*)d_in[3];
  const float* x_proj_w   = (const float*)d_in[4];
  const float* dt_proj_w  = (const float*)d_in[5];
  const float* dt_bias    = (const float*)d_in[6];
  const float* A_logs     = (const float*)d_in[7];
  const float* Ds         = (const float*)d_in[8];
  const float* ln_g       = (const float*)d_in[9];
  const float* ln_b       = (const float*)d_in[10];
  const float* out_proj_w = (const float*)d_in[11];
  float* out = (float*)d_out;

  float* ws     = (float*)d_ws;
  float* x_ssm  = ws;                       //  4,194,304 floats (B,L,Dinner)
  float* z      = ws + 4194304;             //  4,194,304
  float* xc     = ws + 8388608;             //  4,194,304 (B,L,Dinner)
  float* x_dbl  = ws + 12582912;            //  1,572,864 (B,K,L,48)
  float* ysb    = ws + 14155776;            // 16,777,216 (B,K,L,Dinner)
  float* yf     = ws + 30932992;            //  4,194,304 (B,L,Dinner)

  k_in_proj <<<4096, 256, 0, stream>>>(x, in_proj_w, x_ssm, z);
  k_conv    <<<16384, 256, 0, stream>>>(x_ssm, conv_w, conv_b, xc);
  k_xdbl    <<<768, 256, 0, stream>>>(xc, x_proj_w, x_dbl);
  k_scan    <<<BATCH * K_DIR, 512, 0, stream>>>(xc, x_dbl, dt_proj_w, dt_bias,
                                                A_logs, Ds, ysb);
  k_norm    <<<BATCH * SEQ_L, 256, 0, stream>>>(ysb, z, ln_g, ln_b, yf);
  k_out_proj<<<1024, 256, 0, stream>>>(yf, out_proj_w, out);
}